// GCN_77352361001079
// MI455X (gfx1250) — compile-verified
//
#include <hip/hip_runtime.h>
#include <hip/hip_bf16.h>

#define N_NODES  100000
#define N_EDGES  3200000
#define N_GRAPHS 1000
#define FEAT     512
#define EMB      7
#define NCLS     10

typedef __attribute__((ext_vector_type(2))) float v2f;
typedef __attribute__((ext_vector_type(8))) float v8f;

// ---- init: deg=1 (self loop), zero pooled sums/counts --------------------
__global__ void k_init(int* __restrict__ deg, float* __restrict__ sums,
                       int* __restrict__ cnt) {
  int i = blockIdx.x * blockDim.x + threadIdx.x;
  if (i < N_NODES) deg[i] = 1;
  if (i < N_GRAPHS * EMB) sums[i] = 0.0f;
  if (i < N_GRAPHS) cnt[i] = 0;
}

// ---- degree count over dst ------------------------------------------------
__global__ void k_deg(const int* __restrict__ dst, int* __restrict__ deg) {
  int e = blockIdx.x * blockDim.x + threadIdx.x;
  if (e < N_EDGES) atomicAdd(&deg[dst[e]], 1);
}

__global__ void k_dinv(const int* __restrict__ deg, float* __restrict__ dinv) {
  int n = blockIdx.x * blockDim.x + threadIdx.x;
  if (n < N_NODES) dinv[n] = rsqrtf((float)deg[n]);
}

// ---- h = x @ W_in via V_WMMA_F32_16X16X4_F32; hn = h*dinv; agg = hn ------
// One wave handles 16 nodes. A: 16x4 tile of x; B: 4x16 of W (EMB padded to
// 16 with zeros in LDS); C: 16x16 f32 accumulator (8 VGPRs).
__global__ void __launch_bounds__(256) k_gemm(const float* __restrict__ x,
                                              const float* __restrict__ W,
                                              const float* __restrict__ dinv,
                                              float* __restrict__ hn,
                                              float* __restrict__ agg) {
  __shared__ float Wp[FEAT * 16];
  for (int i = threadIdx.x; i < FEAT * 16; i += blockDim.x) {
    int r = i >> 4, c2 = i & 15;
    Wp[i] = (c2 < EMB) ? W[r * EMB + c2] : 0.0f;
  }
  __syncthreads();

  int wave = threadIdx.x >> 5;
  int lane = threadIdx.x & 31;
  int tile = blockIdx.x * (blockDim.x >> 5) + wave;
  int node0 = tile * 16;
  if (node0 >= N_NODES) return;   // uniform per wave -> EXEC all-1 below

  int mn   = lane & 15;           // M index for A, N index for B/C/D
  int koff = (lane >> 4) << 1;    // lane half selects K {0,1} vs {2,3}

  const float* xrow = x + (size_t)(node0 + mn) * FEAT;
  v8f c = {0.f, 0.f, 0.f, 0.f, 0.f, 0.f, 0.f, 0.f};

  for (int k = 0; k < FEAT; k += 4) {
    // A 16x4: vgpr0 = K=koff, vgpr1 = K=koff+1 (8B-aligned float2 load)
    v2f a = *(const v2f*)(xrow + k + koff);
    // B 4x16 mirrored: vgpr0 = B[k+koff][n], vgpr1 = B[k+koff+1][n]
    v2f b;
    b.x = Wp[(k + koff) * 16 + mn];
    b.y = Wp[(k + koff + 1) * 16 + mn];
    c = __builtin_amdgcn_wmma_f32_16x16x4_f32(false, a, false, b,
                                              (short)0, c, false, false);
  }

  // C/D: VGPR r -> M = r (lanes 0-15) or 8+r (lanes 16-31), N = lane&15
  int mbase = (lane >> 4) * 8;
  if (mn < EMB) {
#pragma unroll
    for (int r = 0; r < 8; ++r) {
      int node = node0 + mbase + r;
      float v = c[r] * dinv[node];
      hn[node * EMB + mn]  = v;   // scaled messages for scatter
      agg[node * EMB + mn] = v;   // self-loop contribution seeds agg
    }
  }
}

// ---- edge scatter: agg[dst] += hn[src] (L2-resident fp32 atomics) --------
__global__ void k_scatter(const int* __restrict__ src, const int* __restrict__ dst,
                          const float* __restrict__ hn, float* __restrict__ agg) {
  int e = blockIdx.x * blockDim.x + threadIdx.x;
  if (e >= N_EDGES) return;
  int s = src[e], d = dst[e];
  const float* hs = hn + (size_t)s * EMB;
  float* ad = agg + (size_t)d * EMB;
#pragma unroll
  for (int j = 0; j < EMB; ++j) unsafeAtomicAdd(&ad[j], hs[j]);
}

// ---- per node: out = relu(dinv*agg + b_in); pool into graph sums ---------
__global__ void k_node(const float* __restrict__ agg, const float* __restrict__ dinv,
                       const float* __restrict__ b_in, const int* __restrict__ batch,
                       float* __restrict__ sums, int* __restrict__ cnt) {
  int n = blockIdx.x * blockDim.x + threadIdx.x;
  if (n >= N_NODES) return;
  int g = batch[n];
  float di = dinv[n];
#pragma unroll
  for (int j = 0; j < EMB; ++j) {
    float v = fmaxf(di * agg[n * EMB + j] + b_in[j], 0.0f);
    unsafeAtomicAdd(&sums[g * EMB + j], v);
  }
  atomicAdd(&cnt[g], 1);
}

// ---- per graph: mean pool, logits = pooled@W_out + b_out, softmax --------
__global__ void k_head(const float* __restrict__ sums, const int* __restrict__ cnt,
                       const float* __restrict__ Wout, const float* __restrict__ bout,
                       float* __restrict__ out) {
  int g = blockIdx.x * blockDim.x + threadIdx.x;
  if (g >= N_GRAPHS) return;
  float inv = 1.0f / fmaxf((float)cnt[g], 1.0f);
  float p[EMB];
#pragma unroll
  for (int j = 0; j < EMB; ++j) p[j] = sums[g * EMB + j] * inv;
  float l[NCLS];
  float m = -3.4e38f;
#pragma unroll
  for (int k = 0; k < NCLS; ++k) {
    float acc = bout[k];
#pragma unroll
    for (int j = 0; j < EMB; ++j) acc += p[j] * Wout[j * NCLS + k];
    l[k] = acc;
    m = fmaxf(m, acc);
  }
  float s = 0.0f;
#pragma unroll
  for (int k = 0; k < NCLS; ++k) { l[k] = __expf(l[k] - m); s += l[k]; }
  float is = 1.0f / s;
#pragma unroll
  for (int k = 0; k < NCLS; ++k) out[g * NCLS + k] = l[k] * is;
}

extern "C" void kernel_launch(void* const* d_in, const int* in_sizes, int n_in,
                              void* d_out, int out_size, void* d_ws, size_t ws_size,
                              hipStream_t stream) {
  (void)in_sizes; (void)n_in; (void)out_size; (void)ws_size;
  const float* x     = (const float*)d_in[0];
  const int*   ei    = (const int*)d_in[1];   // [2, E]: src then dst
  const int*   batch = (const int*)d_in[2];
  const float* W_in  = (const float*)d_in[3];
  const float* b_in  = (const float*)d_in[4];
  // d_in[5]/d_in[6] (W_conv1/b_conv1) are dead w.r.t. the returned logits.
  const float* W_out = (const float*)d_in[7];
  const float* b_out = (const float*)d_in[8];
  float* out = (float*)d_out;

  const int* src = ei;
  const int* dst = ei + N_EDGES;

  // Workspace layout (≈6.44 MB total)
  char*  ws   = (char*)d_ws;
  int*   deg  = (int*)(ws);                // 400,000 B
  float* dinv = (float*)(ws + 400000);     // 400,000 B
  float* hn   = (float*)(ws + 800000);     // 2,800,000 B
  float* agg  = (float*)(ws + 3600000);    // 2,800,000 B
  float* sums = (float*)(ws + 6400000);    // 28,000 B
  int*   cnt  = (int*)(ws + 6428000);      // 4,000 B

  k_init<<<(N_NODES + 255) / 256, 256, 0, stream>>>(deg, sums, cnt);
  k_deg<<<(N_EDGES + 255) / 256, 256, 0, stream>>>(dst, deg);
  k_dinv<<<(N_NODES + 255) / 256, 256, 0, stream>>>(deg, dinv);
  int tiles = (N_NODES + 15) / 16;                 // 6250
  k_gemm<<<(tiles + 7) / 8, 256, 0, stream>>>(x, W_in, dinv, hn, agg);
  k_scatter<<<(N_EDGES + 255) / 256, 256, 0, stream>>>(src, dst, hn, agg);
  k_node<<<(N_NODES + 255) / 256, 256, 0, stream>>>(agg, dinv, b_in, batch, sums, cnt);
  k_head<<<(N_GRAPHS + 255) / 256, 256, 0, stream>>>(sums, cnt, W_out, b_out, out);
}